// GHMC_24644522344916
// MI455X (gfx1250) — compile-verified
//
#include <hip/hip_runtime.h>
#include <hip/hip_bf16.h>

// GHM-C loss, single-pass packed-histogram formulation.
// loss = (1/n) * sum_b bce_sum[b] / max(count[b],1),  n = #nonempty bins.
// (tot cancels exactly between w's numerator and the final /tot.)
//
// Per element: ONE ds_add_f64 of (bce + 2^32) -> count accumulates in the high
// mantissa bits, bce sum in the low ones. Per-wave sub-histograms keep
// magnitudes < 2^45 so bce keeps >= 8 effective fraction bits.
// Channel loop is unrolled x2 with both b128 loads issued up front so each
// wave keeps 2 cachelines in flight (covers HBM latency at 23.3 TB/s).

#define GHMC_BINS 100
#define GHMC_C    19
#define GHMC_HW   (512 * 512)      // H*W = 2^18
#define GHMC_NPIX (8 * GHMC_HW)    // B*H*W = 2,097,152
#define GHMC_NGRP (GHMC_NPIX / 4)  // float4 pixel groups = 524,288
#define GHMC_TPB  256
#define GHMC_WAVES (GHMC_TPB / 32)
#define GHMC_BLOCKS 1024

#define TWO32   4294967296.0
#define INV232  2.3283064365386963e-10

typedef float v4f __attribute__((ext_vector_type(4)));

__global__ void ghmc_zero_ws(float* ws) {
    int i = threadIdx.x;
    if (i < 2 * GHMC_BINS) ws[i] = 0.0f;
}

__global__ __launch_bounds__(GHMC_TPB) void ghmc_hist(
    const float* __restrict__ pred,
    const long long* __restrict__ target,
    const unsigned char* __restrict__ lw,
    float* __restrict__ g_cnt,   // [BINS]
    float* __restrict__ g_bce)   // [BINS]
{
    // wave32-private packed histograms (count*2^32 + bce_sum), 6.4 KB of LDS
    __shared__ double s_hist[GHMC_WAVES][GHMC_BINS];

    const int tid  = threadIdx.x;
    const int wave = tid >> 5;

    for (int i = tid; i < GHMC_WAVES * GHMC_BINS; i += GHMC_TPB)
        (&s_hist[0][0])[i] = 0.0;
    __syncthreads();

    const int nthreads = GHMC_BLOCKS * GHMC_TPB;
    for (int g = blockIdx.x * GHMC_TPB + tid; g < GHMC_NGRP; g += nthreads) {
        const int pix4 = g << 2;                 // first of 4 consecutive pixels
        const int b    = pix4 >> 18;             // / HW
        const int hw   = pix4 & (GHMC_HW - 1);   // % HW

        // per-pixel metadata loaded ONCE, reused for all 19 channels
        longlong2 t01 = *(const longlong2*)(target + pix4);
        longlong2 t23 = *(const longlong2*)(target + pix4 + 2);
        uchar4    m4  = *(const uchar4*)(lw + pix4);
        int  tc[4] = { (int)t01.x, (int)t01.y, (int)t23.x, (int)t23.y };
        bool v[4]  = { m4.x != 0, m4.y != 0, m4.z != 0, m4.w != 0 };

        const float* pbase = pred + (((size_t)(b * GHMC_C)) << 18) + hw;

        // process one channel's float4 (4 pixels)
        auto process = [&](const v4f x4, const int c) {
            #pragma unroll
            for (int j = 0; j < 4; ++j) {
                if (!v[j]) continue;
                const float x = x4[j];
                // e = exp(-|x|) via v_exp_f32; abs/neg fold into the mul
                const float e = __builtin_amdgcn_exp2f(-fabsf(x) * 1.4426950408889634f);
                const float d = 1.0f + e;
                // u = e/(1+e) = min(sig, 1-sig); v_rcp_f32 (22-bit, plenty for binning)
                const float u = e * __builtin_amdgcn_rcpf(d);
                const bool thit = (c == tc[j]);      // one-hot target
                const bool pos  = (x >= 0.0f);
                // g = |sigmoid(x) - t|
                const float gg = (pos != thit) ? (1.0f - u) : u;
                int bin = (int)(gg * (float)GHMC_BINS);   // g>=0 -> trunc==floor
                bin = bin > (GHMC_BINS - 1) ? (GHMC_BINS - 1) : bin;
                // bce = max(x,0) - x*t + ln(1+e);  ln via v_log_f32 * ln2
                const float bce = fmaf(__builtin_amdgcn_logf(d), 0.6931471805599453f,
                                       fmaxf(x, 0.0f) - (thit ? x : 0.0f));
                // one ds_add_f64: count in bits >=32, bce sum below
                const double packed = (double)bce + TWO32;
                __hip_atomic_fetch_add(&s_hist[wave][bin], packed,
                                       __ATOMIC_RELAXED, __HIP_MEMORY_SCOPE_WORKGROUP);
            }
        };

        // channel loop: pairs with both NT b128 loads in flight before compute
        #pragma unroll 1
        for (int c = 0; c < GHMC_C - 1; c += 2) {
            const v4f xa = __builtin_nontemporal_load(
                (const v4f*)(pbase + ((size_t)c << 18)));
            const v4f xb = __builtin_nontemporal_load(
                (const v4f*)(pbase + ((size_t)(c + 1) << 18)));
            process(xa, c);
            process(xb, c + 1);
        }
        {   // tail channel (C=19 is odd)
            const v4f xt = __builtin_nontemporal_load(
                (const v4f*)(pbase + ((size_t)(GHMC_C - 1) << 18)));
            process(xt, GHMC_C - 1);
        }
    }
    __syncthreads();

    // fold wave histograms, unpack, publish with HW global_atomic_add_f32
    for (int bin = tid; bin < GHMC_BINS; bin += GHMC_TPB) {
        double tot = 0.0;
        #pragma unroll
        for (int w = 0; w < GHMC_WAVES; ++w) tot += s_hist[w][bin];
        const double cnt = floor(tot * INV232);        // exact: cnt*2^32 < 2^52
        const double bs  = fma(-cnt, TWO32, tot);      // exact residual = bce sum
        unsafeAtomicAdd(&g_cnt[bin], (float)cnt);
        unsafeAtomicAdd(&g_bce[bin], (float)bs);
    }
}

__global__ __launch_bounds__(128) void ghmc_final(
    const float* __restrict__ g_cnt,
    const float* __restrict__ g_bce,
    float* __restrict__ out)
{
    __shared__ float s_term[128];
    __shared__ float s_n[128];
    const int tid = threadIdx.x;
    float term = 0.0f, nn = 0.0f;
    if (tid < GHMC_BINS) {
        const float c = g_cnt[tid];
        if (c > 0.0f) {
            nn   = 1.0f;
            term = g_bce[tid] / fmaxf(c, 1.0f);
        }
    }
    s_term[tid] = term;
    s_n[tid]    = nn;
    __syncthreads();
    for (int s = 64; s > 0; s >>= 1) {
        if (tid < s) { s_term[tid] += s_term[tid + s]; s_n[tid] += s_n[tid + s]; }
        __syncthreads();
    }
    if (tid == 0) {
        const float n = s_n[0];
        out[0] = (n > 0.0f) ? (s_term[0] / n) : 0.0f;   // * LOSS_WEIGHT (=1)
    }
}

extern "C" void kernel_launch(void* const* d_in, const int* in_sizes, int n_in,
                              void* d_out, int out_size, void* d_ws, size_t ws_size,
                              hipStream_t stream) {
    const float*         pred   = (const float*)d_in[0];
    const long long*     target = (const long long*)d_in[1];      // int64
    const unsigned char* lw     = (const unsigned char*)d_in[2];  // bool

    float* ws    = (float*)d_ws;     // [0,100) counts, [100,200) bce sums
    float* g_cnt = ws;
    float* g_bce = ws + GHMC_BINS;

    ghmc_zero_ws<<<1, 256, 0, stream>>>(ws);
    ghmc_hist<<<GHMC_BLOCKS, GHMC_TPB, 0, stream>>>(pred, target, lw, g_cnt, g_bce);
    ghmc_final<<<1, 128, 0, stream>>>(g_cnt, g_bce, (float*)d_out);
}